// GATv2NodeClassifier_49744311222478
// MI455X (gfx1250) — compile-verified
//
#include <hip/hip_runtime.h>
#include <math.h>

#define IN_CH   128
#define HHID    128      // HEADS*HID = HEADS*OUT-per-head-total = 128 for both layers
#define NHEADS  8
#define CH      16       // per-head channels (HID == OUT == 16)
#define NEG_SL  0.2f
#define BN_EPS  1e-5f
#define LN_EPS  1e-5f

typedef float v2f __attribute__((ext_vector_type(2)));
typedef float v8f __attribute__((ext_vector_type(8)));

// ---- order-preserving float->uint key for atomic max on signed floats ----
__device__ __forceinline__ unsigned fkey(float f) {
    unsigned u = __float_as_uint(f);
    return (u & 0x80000000u) ? ~u : (u | 0x80000000u);
}
__device__ __forceinline__ float funkey(unsigned k) {
    unsigned u = (k & 0x80000000u) ? (k ^ 0x80000000u) : ~k;
    return __uint_as_float(u);
}

// ============================================================================
// WMMA fp32 GEMM: C[nrows,ocols] = X[nrows,128] @ W[ocols,128]^T (+ bias[col])
// One wave per 16x16 tile. K=128 fixed. v_wmma_f32_16x16x4_f32, 32 steps.
// ============================================================================
__global__ __launch_bounds__(32)
void gemm16_wmma(const float* __restrict__ X, const float* __restrict__ W,
                 const float* __restrict__ bias, float* __restrict__ C,
                 int nrows, int ocols)
{
    __shared__ float lA[16 * 132];   // +4 pad: bank-conflict-free fragment reads
    __shared__ float lW[16 * 132];
    const int tid = threadIdx.x;             // 0..31 (wave32)
    const int m0  = blockIdx.x * 16;
    const int n0  = blockIdx.y * 16;
    const int c4  = tid * 4;                 // each lane stages 4 consecutive floats/row

    #pragma unroll
    for (int i = 0; i < 16; ++i) {
        int r = m0 + i;
        float4 v = make_float4(0.f, 0.f, 0.f, 0.f);
        if (r < nrows) v = *(const float4*)(X + (size_t)r * IN_CH + c4);
        *(float4*)(lA + i * 132 + c4) = v;
        float4 w = *(const float4*)(W + (size_t)(n0 + i) * IN_CH + c4);
        *(float4*)(lW + i * 132 + c4) = w;
    }
    __syncthreads();

    // ISA 16x4 f32 A layout: lanes 0-15 -> K = k,k+1 ; lanes 16-31 -> K = k+2,k+3
    const int half2 = (tid >> 4) * 2;
    const int r     = tid & 15;
    const float* pa = lA + r * 132;          // r = M row for A
    const float* pb = lW + r * 132;          // r = N col for B (row of W)
    v8f acc = {0.f, 0.f, 0.f, 0.f, 0.f, 0.f, 0.f, 0.f};

    #pragma unroll
    for (int kk = 0; kk < IN_CH; kk += 4) {
        const int k = kk + half2;
        v2f a; a.x = pa[k]; a.y = pa[k + 1];
        v2f b; b.x = pb[k]; b.y = pb[k + 1];
        acc = __builtin_amdgcn_wmma_f32_16x16x4_f32(
                  false, a, false, b, (short)0, acc, false, false);
    }

    // C/D layout: VGPR j, lanes 0-15 -> M=j ; lanes 16-31 -> M=j+8 ; N = lane&15
    const float bn = bias ? bias[n0 + r] : 0.f;
    #pragma unroll
    for (int j = 0; j < 8; ++j) {
        int m = m0 + j + ((tid >> 4) << 3);
        if (m < nrows) C[(size_t)m * ocols + (n0 + r)] = acc[j] + bn;
    }
}

// ============================================================================
// Per-node init: zero accumulator, denominators, and smax keys
// ============================================================================
__global__ void init_nodes(float* __restrict__ acc, float* __restrict__ den,
                           unsigned* __restrict__ smax, int n)
{
    int t = blockIdx.x * blockDim.x + threadIdx.x;
    if (t < n * HHID) acc[t] = 0.f;
    if (t < n * NHEADS) { den[t] = 0.f; smax[t] = 0u; }
}

// ============================================================================
// Edge pass 1: score(e,h) = att[h]·leaky_relu(xl[src]+xr[dst]); atomic max/dst
// One thread per (edge, head); self-loops appended at e >= E.
// ============================================================================
__global__ void edge_max_kernel(const int* __restrict__ ei, int E,
                                const float* __restrict__ XL,
                                const float* __restrict__ XR,
                                const float* __restrict__ att,
                                unsigned* __restrict__ smax, int nthreads)
{
    int t = blockIdx.x * blockDim.x + threadIdx.x;
    if (t >= nthreads) return;
    int e = t >> 3, h = t & 7;
    int s, d;
    if (e < E) { s = ei[e]; d = ei[E + e]; } else { s = e - E; d = s; }

    const float4* pl = (const float4*)(XL + (size_t)s * HHID + h * CH);
    const float4* pr = (const float4*)(XR + (size_t)d * HHID + h * CH);
    const float4* pw = (const float4*)(att + h * CH);
    float sc = 0.f;
    #pragma unroll
    for (int i = 0; i < CH / 4; ++i) {
        float4 a = pl[i], b = pr[i], w = pw[i];
        float v;
        v = a.x + b.x; v = v > 0.f ? v : NEG_SL * v; sc += w.x * v;
        v = a.y + b.y; v = v > 0.f ? v : NEG_SL * v; sc += w.y * v;
        v = a.z + b.z; v = v > 0.f ? v : NEG_SL * v; sc += w.z * v;
        v = a.w + b.w; v = v > 0.f ? v : NEG_SL * v; sc += w.w * v;
    }
    atomicMax(&smax[(size_t)d * NHEADS + h], fkey(sc));
}

// ============================================================================
// Edge pass 2: ex = exp(score - max); den[dst,h] += ex; acc[dst,h,:] += ex*xl[src,h,:]
// (normalization by den deferred to the per-node finish kernels)
// ============================================================================
__global__ void edge_accum_kernel(const int* __restrict__ ei, int E,
                                  const float* __restrict__ XL,
                                  const float* __restrict__ XR,
                                  const float* __restrict__ att,
                                  const unsigned* __restrict__ smax,
                                  float* __restrict__ den,
                                  float* __restrict__ acc, int nthreads)
{
    int t = blockIdx.x * blockDim.x + threadIdx.x;
    if (t >= nthreads) return;
    int e = t >> 3, h = t & 7;
    int s, d;
    if (e < E) { s = ei[e]; d = ei[E + e]; } else { s = e - E; d = s; }

    const float4* pl = (const float4*)(XL + (size_t)s * HHID + h * CH);
    const float4* pr = (const float4*)(XR + (size_t)d * HHID + h * CH);
    const float4* pw = (const float4*)(att + h * CH);
    float4 l[CH / 4];
    float sc = 0.f;
    #pragma unroll
    for (int i = 0; i < CH / 4; ++i) {
        float4 a = pl[i], b = pr[i], w = pw[i];
        l[i] = a;
        float v;
        v = a.x + b.x; v = v > 0.f ? v : NEG_SL * v; sc += w.x * v;
        v = a.y + b.y; v = v > 0.f ? v : NEG_SL * v; sc += w.y * v;
        v = a.z + b.z; v = v > 0.f ? v : NEG_SL * v; sc += w.z * v;
        v = a.w + b.w; v = v > 0.f ? v : NEG_SL * v; sc += w.w * v;
    }
    float m  = funkey(smax[(size_t)d * NHEADS + h]);
    float ex = __expf(sc - m);
    atomicAdd(&den[(size_t)d * NHEADS + h], ex);
    float* o = acc + (size_t)d * HHID + h * CH;
    #pragma unroll
    for (int i = 0; i < CH / 4; ++i) {
        atomicAdd(o + i * 4 + 0, ex * l[i].x);
        atomicAdd(o + i * 4 + 1, ex * l[i].y);
        atomicAdd(o + i * 4 + 2, ex * l[i].z);
        atomicAdd(o + i * 4 + 3, ex * l[i].w);
    }
}

// ============================================================================
// Node finish (layer 1): normalize by den, +bias1, BatchNorm(eval), ELU -> H
// ============================================================================
__global__ void finish1_kernel(const float* __restrict__ acc, const float* __restrict__ den,
                               const float* __restrict__ bias1,
                               const float* __restrict__ bw, const float* __restrict__ bb,
                               const float* __restrict__ rm, const float* __restrict__ rv,
                               float* __restrict__ H, int n)
{
    int t = blockIdx.x * blockDim.x + threadIdx.x;
    if (t >= n * HHID) return;
    int node = t >> 7, c = t & 127, h = c >> 4;
    float v = acc[t] / den[(size_t)node * NHEADS + h] + bias1[c];
    v = (v - rm[c]) * rsqrtf(rv[c] + BN_EPS) * bw[c] + bb[c];
    v = v > 0.f ? v : (__expf(v) - 1.f);   // ELU (alpha=1)
    H[t] = v;
}

// ============================================================================
// Node finish (layer 2): per-head normalize, mean over heads, +bias2, +skip, LayerNorm
// ============================================================================
__global__ void finish2_kernel(const float* __restrict__ acc, const float* __restrict__ den,
                               const float* __restrict__ bias2, const float* __restrict__ xres,
                               const float* __restrict__ lnw, const float* __restrict__ lnb,
                               float* __restrict__ out, int n)
{
    int node = blockIdx.x * blockDim.x + threadIdx.x;
    if (node >= n) return;
    float dinv[NHEADS];
    #pragma unroll
    for (int h = 0; h < NHEADS; ++h) dinv[h] = 1.f / den[(size_t)node * NHEADS + h];
    float y[CH];
    float mu = 0.f;
    #pragma unroll
    for (int c = 0; c < CH; ++c) {
        float s = 0.f;
        #pragma unroll
        for (int h = 0; h < NHEADS; ++h)
            s += acc[(size_t)node * HHID + h * CH + c] * dinv[h];
        float v = s * 0.125f + bias2[c] + xres[(size_t)node * CH + c];
        y[c] = v; mu += v;
    }
    mu *= (1.f / CH);
    float var = 0.f;
    #pragma unroll
    for (int c = 0; c < CH; ++c) { float dv = y[c] - mu; var += dv * dv; }
    var *= (1.f / CH);
    float rs = rsqrtf(var + LN_EPS);
    #pragma unroll
    for (int c = 0; c < CH; ++c)
        out[(size_t)node * CH + c] = (y[c] - mu) * rs * lnw[c] + lnb[c];
}

// ============================================================================
extern "C" void kernel_launch(void* const* d_in, const int* in_sizes, int n_in,
                              void* d_out, int out_size, void* d_ws, size_t ws_size,
                              hipStream_t stream)
{
    const float* x     = (const float*)d_in[0];
    const int*   ei    = (const int*)  d_in[1];
    const float* Wl1   = (const float*)d_in[2];
    const float* bl1   = (const float*)d_in[3];
    const float* Wr1   = (const float*)d_in[4];
    const float* br1   = (const float*)d_in[5];
    const float* att1  = (const float*)d_in[6];
    const float* bias1 = (const float*)d_in[7];
    const float* bn_w  = (const float*)d_in[8];
    const float* bn_b  = (const float*)d_in[9];
    const float* bn_rm = (const float*)d_in[10];
    const float* bn_rv = (const float*)d_in[11];
    const float* Wl2   = (const float*)d_in[12];
    const float* bl2   = (const float*)d_in[13];
    const float* Wr2   = (const float*)d_in[14];
    const float* br2   = (const float*)d_in[15];
    const float* att2  = (const float*)d_in[16];
    const float* bias2 = (const float*)d_in[17];
    const float* Wskip = (const float*)d_in[18];
    const float* ln_w  = (const float*)d_in[19];
    const float* ln_b  = (const float*)d_in[20];

    const int N    = in_sizes[0] / IN_CH;   // 50000
    const int E    = in_sizes[1] / 2;       // 800000
    const int Etot = E + N;                 // + self-loops

    // ---- workspace carve (floats) ----
    float* ws = (float*)d_ws;
    size_t o = 0;
    float*    XL   = ws + o; o += (size_t)N * HHID;
    float*    XR   = ws + o; o += (size_t)N * HHID;
    float*    H    = ws + o; o += (size_t)N * HHID;
    float*    ACC  = ws + o; o += (size_t)N * HHID;
    float*    XRES = ws + o; o += (size_t)N * CH;
    float*    DEN  = ws + o; o += (size_t)N * NHEADS;
    unsigned* SMAX = (unsigned*)(ws + o);

    const int mtiles = (N + 15) / 16;
    dim3 gb(mtiles, HHID / 16);    // 128-wide GEMMs
    dim3 gs(mtiles, 1);            // skip GEMM (16-wide)
    const int nodeT = N * HHID;
    const int edgeT = Etot * NHEADS;
    const int TB = 256;

    // ---------------- layer 1 ----------------
    gemm16_wmma<<<gb, 32, 0, stream>>>(x, Wl1, bl1, XL, N, HHID);
    gemm16_wmma<<<gb, 32, 0, stream>>>(x, Wr1, br1, XR, N, HHID);
    gemm16_wmma<<<gs, 32, 0, stream>>>(x, Wskip, nullptr, XRES, N, CH);

    init_nodes<<<(nodeT + TB - 1) / TB, TB, 0, stream>>>(ACC, DEN, SMAX, N);
    edge_max_kernel  <<<(edgeT + TB - 1) / TB, TB, 0, stream>>>(ei, E, XL, XR, att1, SMAX, edgeT);
    edge_accum_kernel<<<(edgeT + TB - 1) / TB, TB, 0, stream>>>(ei, E, XL, XR, att1, SMAX, DEN, ACC, edgeT);
    finish1_kernel<<<(nodeT + TB - 1) / TB, TB, 0, stream>>>(ACC, DEN, bias1, bn_w, bn_b, bn_rm, bn_rv, H, N);

    // ---------------- layer 2 ----------------
    gemm16_wmma<<<gb, 32, 0, stream>>>(H, Wl2, bl2, XL, N, HHID);
    gemm16_wmma<<<gb, 32, 0, stream>>>(H, Wr2, br2, XR, N, HHID);

    init_nodes<<<(nodeT + TB - 1) / TB, TB, 0, stream>>>(ACC, DEN, SMAX, N);
    edge_max_kernel  <<<(edgeT + TB - 1) / TB, TB, 0, stream>>>(ei, E, XL, XR, att2, SMAX, edgeT);
    edge_accum_kernel<<<(edgeT + TB - 1) / TB, TB, 0, stream>>>(ei, E, XL, XR, att2, SMAX, DEN, ACC, edgeT);
    finish2_kernel<<<(N + TB - 1) / TB, TB, 0, stream>>>(ACC, DEN, bias2, XRES, ln_w, ln_b, (float*)d_out, N);
}